// ResidualQuantizer_60928406061059
// MI455X (gfx1250) — compile-verified
//
#include <hip/hip_runtime.h>

typedef float v2f __attribute__((ext_vector_type(2)));
typedef float v8f __attribute__((ext_vector_type(8)));

#define DIMS 64
#define TILE_K 16
#define CHUNK_CODES 128
#define WAVES_PER_BLOCK 8
#define ROWS_PER_WAVE 16
#define ROWS_PER_BLOCK (WAVES_PER_BLOCK * ROWS_PER_WAVE)

// 32-bit LDS byte offset of a __shared__ object (for async-to-LDS VDST operand).
__device__ __forceinline__ unsigned lds_offset(const void* p) {
    return (unsigned)(unsigned long long)(__attribute__((address_space(3))) const void*)p;
}

// ---- Pass 1: c2[k] = ||codebook[k]||^2 --------------------------------------
__global__ void vq_c2_kernel(const float* __restrict__ cb,
                             float* __restrict__ c2, int K) {
    int k = blockIdx.x * blockDim.x + threadIdx.x;
    if (k < K) {
        const float4* p = (const float4*)(cb + (size_t)k * DIMS);
        float s = 0.f;
#pragma unroll
        for (int i = 0; i < DIMS / 4; ++i) {
            float4 v = p[i];
            s += v.x * v.x + v.y * v.y + v.z * v.z + v.w * v.w;
        }
        c2[k] = s;
    }
}

// ---- Pass 2: WMMA GEMM + fused argmin + gather ------------------------------
__global__ __launch_bounds__(256)
void vq_assign_kernel(const float* __restrict__ R,
                      const float* __restrict__ Cb,
                      const float* __restrict__ c2,
                      float* __restrict__ quant,
                      int* __restrict__ codes,
                      int N, int K) {
    __shared__ float lds_cb[CHUNK_CODES * DIMS];           // 32 KB
    __shared__ int   lds_code[WAVES_PER_BLOCK][ROWS_PER_WAVE];

    const int tid  = threadIdx.x;
    const int wave = tid >> 5;
    const int lane = tid & 31;
    const int half = lane >> 4;      // 0: lanes 0-15, 1: lanes 16-31
    const int lc   = lane & 15;
    const int rowBase = blockIdx.x * ROWS_PER_BLOCK + wave * ROWS_PER_WAVE;

    // A-matrix (16x4 f32 per WMMA): VGPR v, lane L -> row=L%16, k=v+2*(L/16).
    // Per D-chunk of 4, each lane loads a contiguous float2. Keep all 16 chunks
    // (16 rows x 64 cols) resident in VGPRs for the whole K loop.
    v2f a[16];
    {
        const float* ar = R + (size_t)(rowBase + lc) * DIMS + 2 * half;
#pragma unroll
        for (int ch = 0; ch < 16; ++ch)
            a[ch] = *(const v2f*)(ar + 4 * ch);
    }

    // Running argmin: lane holds 8 rows (M = 8*half + r), one code column/tile.
    float minv[8];
    int   mini[8];
#pragma unroll
    for (int r = 0; r < 8; ++r) { minv[r] = 3.402823466e38f; mini[r] = 0; }

    const unsigned ldsBase = lds_offset(lds_cb) + (unsigned)tid * 16u;

    const int nChunks = K / CHUNK_CODES;
    for (int s = 0; s < nChunks; ++s) {
        // Prefetch the next chunk into L2 (one 128B line per lane = 32KB).
        if (s + 1 < nChunks)
            __builtin_prefetch(Cb + (size_t)(s + 1) * CHUNK_CODES * DIMS + tid * 32, 0, 1);

        // Async-stage 128 codes (32 KB) into LDS: 8 x b128 per thread, no VGPR
        // round trip, all in flight at once (tracked by ASYNCcnt).
        {
            unsigned long long g =
                (unsigned long long)(Cb + (size_t)s * CHUNK_CODES * DIMS) +
                (unsigned long long)tid * 16ull;
#pragma unroll
            for (int i = 0; i < 8; ++i) {
                unsigned l = ldsBase + (unsigned)(i * 4096);
                unsigned long long ga = g + (unsigned long long)(i * 4096);
                asm volatile("global_load_async_to_lds_b128 %0, %1, off"
                             :: "v"(l), "v"(ga) : "memory");
            }
        }

        // Hoist this chunk's c2 values while the async staging is in flight.
        float c2v[CHUNK_CODES / TILE_K];
#pragma unroll
        for (int kt = 0; kt < CHUNK_CODES / TILE_K; ++kt)
            c2v[kt] = c2[s * CHUNK_CODES + kt * TILE_K + lc];

        asm volatile("s_wait_asynccnt 0x0" ::: "memory");
        __syncthreads();

#pragma unroll 1
        for (int kt = 0; kt < CHUNK_CODES / TILE_K; ++kt) {
            const int codeIdx = s * CHUNK_CODES + kt * TILE_K + lc;
            // B-matrix (4x16): VGPR v, lane L -> k=v+2*(L/16), col=L%16.
            const float* br = lds_cb + (kt * TILE_K + lc) * DIMS + 2 * half;

            v8f acc0 = {};
            v8f acc1 = {};
#pragma unroll
            for (int ch = 0; ch < 16; ch += 2) {
                v2f b0 = *(const v2f*)(br + 4 * ch);
                v2f b1 = *(const v2f*)(br + 4 * (ch + 1));
                acc0 = __builtin_amdgcn_wmma_f32_16x16x4_f32(
                    false, a[ch],     false, b0, (short)0, acc0, false, false);
                acc1 = __builtin_amdgcn_wmma_f32_16x16x4_f32(
                    false, a[ch + 1], false, b1, (short)0, acc1, false, false);
            }

            const float cc = c2v[kt];
#pragma unroll
            for (int r = 0; r < 8; ++r) {
                float d = cc - 2.0f * (acc0[r] + acc1[r]);
                if (d < minv[r] || (d == minv[r] && codeIdx < mini[r])) {
                    minv[r] = d; mini[r] = codeIdx;
                }
            }
        }
        __syncthreads();
    }

    // Reduce argmin across the 16 lanes of each half-wave (wave32).
#pragma unroll
    for (int m = 1; m < 16; m <<= 1) {
#pragma unroll
        for (int r = 0; r < 8; ++r) {
            float ov = __shfl_xor(minv[r], m, 16);
            int   oi = __shfl_xor(mini[r], m, 16);
            if (ov < minv[r] || (ov == minv[r] && oi < mini[r])) {
                minv[r] = ov; mini[r] = oi;
            }
        }
    }
    if (lc == 0) {
#pragma unroll
        for (int r = 0; r < 8; ++r)
            lds_code[wave][half * 8 + r] = mini[r];
    }
    __syncthreads();

    // Emit codes.
    if (lane < ROWS_PER_WAVE)
        codes[rowBase + lane] = lds_code[wave][lane];

    // Gather winning codebook rows: 64 floats/row = one float2 per lane.
#pragma unroll 1
    for (int m = 0; m < ROWS_PER_WAVE; ++m) {
        int c = lds_code[wave][m];
        v2f v = ((const v2f*)(Cb + (size_t)c * DIMS))[lane];
        ((v2f*)(quant + (size_t)(rowBase + m) * DIMS))[lane] = v;
    }
}

extern "C" void kernel_launch(void* const* d_in, const int* in_sizes, int n_in,
                              void* d_out, int out_size, void* d_ws, size_t ws_size,
                              hipStream_t stream) {
    const float* residual = (const float*)d_in[0];
    const float* codebook = (const float*)d_in[1];
    const int N = in_sizes[0] / DIMS;   // 262144
    const int K = in_sizes[1] / DIMS;   // 1024

    float* quant = (float*)d_out;
    int*   codes = (int*)((float*)d_out + (size_t)N * DIMS);
    float* c2    = (float*)d_ws;        // K floats of scratch

    vq_c2_kernel<<<(K + 255) / 256, 256, 0, stream>>>(codebook, c2, K);
    vq_assign_kernel<<<N / ROWS_PER_BLOCK, 256, 0, stream>>>(
        residual, codebook, c2, quant, codes, N, K);
}